// LatentAttnMasked_44341242364343
// MI455X (gfx1250) — compile-verified
//
#include <hip/hip_runtime.h>
#include <math.h>

// ---------------------------------------------------------------------------
// Problem constants (from reference)
// ---------------------------------------------------------------------------
static constexpr int B_  = 1024;
static constexpr int DS  = 79800;
static constexpr int DT  = 79800;
static constexpr int KO  = 64;     // n_components (also token count)
static constexpr int E_  = 64;     // embedding dim
static constexpr int A_  = 16;     // attn dim
static constexpr int VD  = 16;     // value dim

// GEMM1 split-K decomposition: DS = SPLIT * NBLK * KBLK exactly.
// DS/KBLK = 1425 = 3*5^2*19 ; SPLIT=75 -> NBLK=19, 2400 waves (~2.3/SIMD).
static constexpr int SPLIT = 75;
static constexpr int KBLK  = 56;                  // K per LDS stage
static constexpr int NBLK  = DS / (SPLIT * KBLK); // 19
static constexpr int CHUNK = NBLK * KBLK;         // 1064 per split
static constexpr int APAD  = 60;                  // LDS row stride (floats)

typedef __attribute__((ext_vector_type(2))) float v2f;
typedef __attribute__((ext_vector_type(4))) float v4f;
typedef __attribute__((ext_vector_type(8))) float v8f;
typedef __attribute__((ext_vector_type(4))) int   v4i;

// ---------------------------------------------------------------------------
// CDNA5 async global->LDS support (probe-guarded; sync fallback compiles too)
// Builtin signature (from clang diagnostic): (v4i AS1*, v4i AS3*, Imm, Imm)
// ---------------------------------------------------------------------------
#if defined(__has_builtin)
#if __has_builtin(__builtin_amdgcn_global_load_async_to_lds_b128)
#define HAVE_ASYNC_LDS 1
#endif
#if __has_builtin(__builtin_amdgcn_s_wait_asynccnt)
#define HAVE_WAIT_BUILTIN 1
#endif
#endif

#ifdef HAVE_ASYNC_LDS
typedef __attribute__((address_space(1))) v4i* gv4i_ptr;
typedef __attribute__((address_space(3))) v4i* lv4i_ptr;
#ifdef HAVE_WAIT_BUILTIN
#define WAIT_ASYNC(n) __builtin_amdgcn_s_wait_asynccnt(n)
#else
#define WAIT_ASYNC(n) asm volatile("s_wait_asynccnt " #n ::: "memory")
#endif
#else
#define WAIT_ASYNC(n) ((void)0)
#endif

// ---------------------------------------------------------------------------
// Kernel: zero the split-K accumulation scratch (atomics accumulate into it).
// ---------------------------------------------------------------------------
__global__ void zero_kernel(float* __restrict__ p, int n) {
    int i = blockIdx.x * blockDim.x + threadIdx.x;
    if (i < n) p[i] = 0.0f;
}

// ---------------------------------------------------------------------------
// Kernel: meanL[k] = sum_d source_mean[d] * source_loadings[d, k]
// ---------------------------------------------------------------------------
__global__ void meanl_kernel(const float* __restrict__ mean,
                             const float* __restrict__ L,
                             float* __restrict__ meanL) {
    __shared__ float red[256];
    const int k = blockIdx.x;
    const int t = threadIdx.x;
    float s = 0.0f;
    for (int d = t; d < DS; d += 256)
        s += mean[d] * L[(size_t)d * KO + k];
    red[t] = s;
    __syncthreads();
    for (int off = 128; off > 0; off >>= 1) {
        if (t < off) red[t] += red[t + off];
        __syncthreads();
    }
    if (t == 0) meanL[k] = red[0];
}

// ---------------------------------------------------------------------------
// Kernel: batch-independent attention precompute (Q and emb@W[1:] bases).
// ---------------------------------------------------------------------------
__global__ void precomp_kernel(const float* __restrict__ sc_emb,
                               const float* __restrict__ fc_emb,
                               const float* __restrict__ W_Q,
                               const float* __restrict__ W_K,
                               const float* __restrict__ W_V,
                               float* __restrict__ Qm,
                               float* __restrict__ scK,
                               float* __restrict__ fcK,
                               float* __restrict__ scV,
                               float* __restrict__ fcV) {
    const int t = blockIdx.x * blockDim.x + threadIdx.x;
    if (t >= KO * A_) return;
    const int q = t >> 4;
    const int a = t & 15;
    float accQ = 0.f, asK = 0.f, afK = 0.f, asV = 0.f, afV = 0.f;
    for (int e = 0; e < E_; ++e) {
        const float se = sc_emb[q * E_ + e];
        const float fe = fc_emb[q * E_ + e];
        const float wq = W_Q[(1 + e) * A_ + a];
        const float wk = W_K[(1 + e) * A_ + a];
        const float wv = W_V[(1 + e) * VD + a];
        accQ += fe * wq;
        asK  += se * wk;
        afK  += fe * wk;
        asV  += se * wv;
        afV  += fe * wv;
    }
    Qm[t]  = accQ;
    scK[t] = asK;
    fcK[t] = afK;
    scV[t] = asV;
    fcV[t] = afV;
}

// ---------------------------------------------------------------------------
// GEMM1: xL[B,64] += x[B,DS] @ L[DS,64]   (split-K, f32 WMMA, LDS staging)
// grid = (SPLIT, B/32), block = 32 (one wave; M=32 rows, all 64 output cols).
//
// x tile [32 x KBLK] is staged into LDS via async global->LDS b128 copies
// (2 rows / instruction, fully coalesced 224B segments), double-buffered on
// ASYNCcnt. A-fragments are then ds_load_b64 from LDS; B (source_loadings,
// 20MB, L2-resident) is loaded directly.
// ---------------------------------------------------------------------------
__device__ __forceinline__ void stage_tile(const float* __restrict__ x,
                                           int row0, int kb,
                                           float* lbuf, int lane) {
    if (lane < 28) {
        const int rr  = lane / 14;        // 0..1 : row within pair
        const int sub = (lane % 14) * 4;  // float offset within row (16B units)
        #pragma unroll
        for (int r = 0; r < 32; r += 2) {
            const float* gsrc = x + (size_t)(row0 + r + rr) * DS + kb + sub;
            float*       ldst = lbuf + (r + rr) * APAD + sub;
#ifdef HAVE_ASYNC_LDS
            float* gnc = const_cast<float*>(gsrc);
            __builtin_amdgcn_global_load_async_to_lds_b128(
                (gv4i_ptr)gnc, (lv4i_ptr)ldst, 0, 0);
#else
            *(v4f*)ldst = *(const v4f*)gsrc;
#endif
        }
    }
}

__global__ void gemm1_kernel(const float* __restrict__ x,
                             const float* __restrict__ L,
                             float* __restrict__ xL) {
    __shared__ float bufA[2][32][APAD];

    const int split = blockIdx.x;
    const int row0  = blockIdx.y * 32;
    const int lane  = threadIdx.x;
    const int half  = lane >> 4;
    const int lr    = lane & 15;

    const int kbase = split * CHUNK;

    v8f acc[2][4];
    #pragma unroll
    for (int m = 0; m < 2; ++m)
        #pragma unroll
        for (int c = 0; c < 4; ++c) acc[m][c] = (v8f){};

    // prologue: stage block 0
    stage_tile(x, row0, kbase, &bufA[0][0][0], lane);

    for (int blk = 0; blk < NBLK; ++blk) {
        const int cur = blk & 1;
        if (blk + 1 < NBLK) {
            stage_tile(x, row0, kbase + (blk + 1) * KBLK,
                       &bufA[cur ^ 1][0][0], lane);
            WAIT_ASYNC(16);   // in-order: block `blk`'s 16 copies complete
        } else {
            WAIT_ASYNC(0);    // drain
        }
        asm volatile("" ::: "memory");

        const int kb = kbase + blk * KBLK;
        #pragma unroll 2
        for (int s = 0; s < KBLK; s += 4) {
            const int kr = s + 2 * half;
            const v2f a0 = *(const v2f*)&bufA[cur][lr][kr];
            const v2f a1 = *(const v2f*)&bufA[cur][16 + lr][kr];

            const float* Lp = L + (size_t)(kb + kr) * KO + lr;
            v2f b0, b1, b2, b3;
            b0.x = Lp[0];   b0.y = Lp[KO];
            b1.x = Lp[16];  b1.y = Lp[KO + 16];
            b2.x = Lp[32];  b2.y = Lp[KO + 32];
            b3.x = Lp[48];  b3.y = Lp[KO + 48];

            acc[0][0] = __builtin_amdgcn_wmma_f32_16x16x4_f32(false, a0, false, b0, (short)0, acc[0][0], false, false);
            acc[0][1] = __builtin_amdgcn_wmma_f32_16x16x4_f32(false, a0, false, b1, (short)0, acc[0][1], false, false);
            acc[0][2] = __builtin_amdgcn_wmma_f32_16x16x4_f32(false, a0, false, b2, (short)0, acc[0][2], false, false);
            acc[0][3] = __builtin_amdgcn_wmma_f32_16x16x4_f32(false, a0, false, b3, (short)0, acc[0][3], false, false);
            acc[1][0] = __builtin_amdgcn_wmma_f32_16x16x4_f32(false, a1, false, b0, (short)0, acc[1][0], false, false);
            acc[1][1] = __builtin_amdgcn_wmma_f32_16x16x4_f32(false, a1, false, b1, (short)0, acc[1][1], false, false);
            acc[1][2] = __builtin_amdgcn_wmma_f32_16x16x4_f32(false, a1, false, b2, (short)0, acc[1][2], false, false);
            acc[1][3] = __builtin_amdgcn_wmma_f32_16x16x4_f32(false, a1, false, b3, (short)0, acc[1][3], false, false);
        }
    }

    #pragma unroll
    for (int m = 0; m < 2; ++m) {
        #pragma unroll
        for (int v = 0; v < 8; ++v) {
            const int row = row0 + m * 16 + v + 8 * half;
            float* dst = xL + (size_t)row * KO + lr;
            atomicAdd(dst + 0,  acc[m][0][v]);
            atomicAdd(dst + 16, acc[m][1][v]);
            atomicAdd(dst + 32, acc[m][2][v]);
            atomicAdd(dst + 48, acc[m][3][v]);
        }
    }
}

// ---------------------------------------------------------------------------
// Masked latent attention epilogue. One 64-thread block per batch item.
// ---------------------------------------------------------------------------
__global__ void attn_kernel(const float* __restrict__ xL,
                            const float* __restrict__ meanL,
                            const float* __restrict__ fc,
                            const int*   __restrict__ mask,
                            const float* __restrict__ Qm,
                            const float* __restrict__ scK,
                            const float* __restrict__ fcK,
                            const float* __restrict__ scV,
                            const float* __restrict__ fcV,
                            const float* __restrict__ W_K,
                            const float* __restrict__ W_V,
                            const float* __restrict__ rW,
                            const float* __restrict__ rb,
                            float* __restrict__ pred) {
    __shared__ float sKm[2 * KO * A_];
    __shared__ float sV [2 * KO * VD];
    __shared__ float sS [2 * KO];
    __shared__ float sVis[KO];

    const int b = blockIdx.x;
    const int t = threadIdx.x;          // 0..63

    const int   mt  = mask[b * KO + t];
    const float fct = fc[b * KO + t];
    const float vis = (mt == 0) ? 1.0f : 0.0f;

    sVis[t]    = vis;
    sS[t]      = xL[b * KO + t] - meanL[t];
    sS[t + KO] = fct * vis;
    __syncthreads();

    for (int j = t; j < 2 * KO; j += 64) {
        const float s   = sS[j];
        const float* bK = (j < KO) ? &scK[j * A_] : &fcK[(j - KO) * A_];
        const float* bV = (j < KO) ? &scV[j * VD] : &fcV[(j - KO) * VD];
        #pragma unroll
        for (int a = 0; a < A_; ++a) {
            sKm[j * A_ + a] = s * W_K[a] + bK[a];
            sV [j * VD + a] = s * W_V[a] + bV[a];
        }
    }
    __syncthreads();

    float q[A_];
    #pragma unroll
    for (int a = 0; a < A_; ++a) q[a] = Qm[t * A_ + a];

    float logits[2 * KO];
    float mx = -3.0e38f;
    for (int j = 0; j < 2 * KO; ++j) {
        float l = 0.0f;
        #pragma unroll
        for (int a = 0; a < A_; ++a) l += q[a] * sKm[j * A_ + a];
        l *= 0.25f;
        if (j >= KO && sVis[j - KO] <= 0.0f) l = -1.0e9f;
        logits[j] = l;
        mx = fmaxf(mx, l);
    }
    float sum = 0.0f;
    for (int j = 0; j < 2 * KO; ++j) {
        const float p = __expf(logits[j] - mx);
        logits[j] = p;
        sum += p;
    }
    const float inv = 1.0f / sum;

    float ctx[VD];
    #pragma unroll
    for (int v = 0; v < VD; ++v) ctx[v] = 0.0f;
    for (int j = 0; j < 2 * KO; ++j) {
        const float p = logits[j];
        #pragma unroll
        for (int v = 0; v < VD; ++v) ctx[v] += p * sV[j * VD + v];
    }

    float r = rb[0];
    #pragma unroll
    for (int v = 0; v < VD; ++v) r += ctx[v] * inv * rW[v];

    pred[b * KO + t] = (mt > 0) ? r : fct;
}

// ---------------------------------------------------------------------------
// Decode GEMM: out[B,DT] = pred[B,64] @ Lt^T + tmean  (f32 WMMA)
// grid = (ceil(DT/64), B/16), block = 32. One wave handles 4 col tiles,
// loading the pred A-fragments once (64 WMMA per wave). Nontemporal stores
// keep the 327MB output stream from evicting L2-resident target_loadings.
// ---------------------------------------------------------------------------
__global__ void decode_kernel(const float* __restrict__ pred,
                              const float* __restrict__ Lt,
                              const float* __restrict__ tmean,
                              float* __restrict__ out) {
    const int col_base = blockIdx.x * 64;
    const int row0     = blockIdx.y * 16;
    const int lane     = threadIdx.x;
    const int half     = lane >> 4;
    const int lr       = lane & 15;

    const float* prow = pred + (size_t)(row0 + lr) * KO;
    v2f a[16];
    #pragma unroll
    for (int kk = 0; kk < KO; kk += 4)
        a[kk >> 2] = *(const v2f*)(prow + kk + 2 * half);

    #pragma unroll
    for (int c = 0; c < 4; ++c) {
        const int  col = col_base + c * 16 + lr;
        const bool ok  = col < DT;
        const float* lcol = Lt + (size_t)(ok ? col : 0) * KO;

        v8f acc = {};
        #pragma unroll
        for (int kk = 0; kk < KO; kk += 4) {
            const int kr = kk + 2 * half;
            const v2f b = ok ? *(const v2f*)(lcol + kr) : (v2f){0.0f, 0.0f};
            acc = __builtin_amdgcn_wmma_f32_16x16x4_f32(false, a[kk >> 2], false, b,
                                                        (short)0, acc, false, false);
        }

        if (ok) {
            const float tm = tmean[col];
            #pragma unroll
            for (int v = 0; v < 8; ++v)
                __builtin_nontemporal_store(
                    acc[v] + tm,
                    out + (size_t)(row0 + v + 8 * half) * DT + col);
        }
    }
}

// ---------------------------------------------------------------------------
// Host launcher
// ---------------------------------------------------------------------------
extern "C" void kernel_launch(void* const* d_in, const int* in_sizes, int n_in,
                              void* d_out, int out_size, void* d_ws, size_t ws_size,
                              hipStream_t stream) {
    const float* x_src           = (const float*)d_in[0];
    const float* fc_scores       = (const float*)d_in[1];
    const int*   mask            = (const int*)  d_in[2];
    const float* source_mean     = (const float*)d_in[3];
    const float* source_loadings = (const float*)d_in[4];
    const float* target_mean     = (const float*)d_in[5];
    const float* target_loadings = (const float*)d_in[6];
    const float* sc_emb          = (const float*)d_in[7];
    const float* fc_emb          = (const float*)d_in[8];
    const float* W_Q             = (const float*)d_in[9];
    const float* W_K             = (const float*)d_in[10];
    const float* W_V             = (const float*)d_in[11];
    const float* readout_W       = (const float*)d_in[12];
    const float* readout_b       = (const float*)d_in[13];

    float* out = (float*)d_out;

    float* ws    = (float*)d_ws;
    float* xL    = ws;                       // B*KO
    float* meanL = xL    + B_ * KO;          // KO
    float* Qm    = meanL + KO;               // KO*A
    float* scK   = Qm    + KO * A_;
    float* fcK   = scK   + KO * A_;
    float* scV   = fcK   + KO * A_;
    float* fcV   = scV   + KO * VD;
    float* pred  = fcV   + KO * VD;          // B*KO
    (void)ws_size; (void)n_in; (void)in_sizes; (void)out_size;

    zero_kernel<<<(B_ * KO + 255) / 256, 256, 0, stream>>>(xL, B_ * KO);

    meanl_kernel<<<KO, 256, 0, stream>>>(source_mean, source_loadings, meanL);

    precomp_kernel<<<4, 256, 0, stream>>>(sc_emb, fc_emb, W_Q, W_K, W_V,
                                          Qm, scK, fcK, scV, fcV);

    gemm1_kernel<<<dim3(SPLIT, B_ / 32), 32, 0, stream>>>(x_src, source_loadings, xL);

    attn_kernel<<<B_, 64, 0, stream>>>(xL, meanL, fc_scores, mask,
                                       Qm, scK, fcK, scV, fcV,
                                       W_K, W_V, readout_W, readout_b, pred);

    decode_kernel<<<dim3((DT + 63) / 64, B_ / 16), 32, 0, stream>>>(
        pred, target_loadings, target_mean, out);
}